// Self_Attention_59502476919448
// MI455X (gfx1250) — compile-verified
//
#include <hip/hip_runtime.h>
#include <math.h>

#define D_MODEL 1024
#define D_HEAD  64
#define NHEADS  16
#define BATCH   4
#define SEQ     2048
#define NROWS   (BATCH * SEQ)   // 8192

typedef float v2f __attribute__((ext_vector_type(2)));
typedef float v8f __attribute__((ext_vector_type(8)));

// V_WMMA_F32_16X16X4_F32: D(16x16 f32) = A(16x4 f32) * B(4x16 f32) + C
// 8-arg pattern: (neg_a, A, neg_b, B, c_mod, C, reuse_a, reuse_b)
static __device__ __forceinline__ v8f wmma_k4(v2f a, v2f b, v8f c) {
  return __builtin_amdgcn_wmma_f32_16x16x4_f32(false, a, false, b, (short)0, c,
                                               false, false);
}

// ---------------------------------------------------------------------------
// Wsum[d, j] = sum_h Wo[h*64 + d, j]   (collapses the tiled output GEMM 16x)
// ---------------------------------------------------------------------------
__global__ void k_wsum(const float* __restrict__ Wo, float* __restrict__ Wsum) {
  int i = blockIdx.x * blockDim.x + threadIdx.x;  // 0 .. 64*1024-1
  int d = i >> 10;
  int j = i & 1023;
  float s = 0.f;
#pragma unroll
  for (int h = 0; h < NHEADS; ++h) s += Wo[(h * D_HEAD + d) * D_MODEL + j];
  Wsum[i] = s;
}

// ---------------------------------------------------------------------------
// Projection: out[n, 0:64] = (X[n, :] @ W + b) * scale   (scale folds 1/sqrt(dh))
// One wave computes a 16(M) x 64(N) tile; K=1024 in steps of 4 via f32 WMMA.
// ---------------------------------------------------------------------------
__global__ void k_proj(const float* __restrict__ X, const float* __restrict__ W,
                       const float* __restrict__ bias, float* __restrict__ out,
                       float scale) {
  const int w    = threadIdx.x >> 5;
  const int lane = threadIdx.x & 31;
  const int col  = lane & 15;
  const int half = lane >> 4;
  const int mbase = (blockIdx.x * 8 + w) * 16;

  const float* xrow = X + (size_t)(mbase + col) * D_MODEL;
  v8f acc[4] = {{}, {}, {}, {}};

  for (int k = 0; k < D_MODEL; k += 4) {
    v2f a;
    a.x = xrow[k + 2 * half];
    a.y = xrow[k + 2 * half + 1];
    const float* w0 = W + (size_t)(k + 2 * half) * D_HEAD + col;
    const float* w1 = w0 + D_HEAD;
#pragma unroll
    for (int t = 0; t < 4; ++t) {
      v2f b;
      b.x = w0[t * 16];
      b.y = w1[t * 16];
      acc[t] = wmma_k4(a, b, acc[t]);
    }
  }
#pragma unroll
  for (int t = 0; t < 4; ++t) {
    float bb = bias[t * 16 + col];
#pragma unroll
    for (int vv = 0; vv < 8; ++vv) {
      int r = mbase + vv + 8 * half;
      out[(size_t)r * D_HEAD + t * 16 + col] = (acc[t][vv] + bb) * scale;
    }
  }
}

// ---------------------------------------------------------------------------
// Pass 1: column softmax stats. softmax is over the QUERY axis, so for each
// key column kc: m[kc] = max_q s(q,kc), l[kc] = sum_q exp(s-m).
// Block = (batch, 16-key-column strip); 8 waves split the 2048 query rows.
// ---------------------------------------------------------------------------
__global__ void k_stats(const float* __restrict__ qb_all,
                        const float* __restrict__ kb_all,
                        float* __restrict__ mOut, float* __restrict__ lOut) {
  __shared__ float sm[8][16], sl[8][16];
  const int w    = threadIdx.x >> 5;
  const int lane = threadIdx.x & 31;
  const int col  = lane & 15;
  const int half = lane >> 4;
  const int ktile = blockIdx.x & (SEQ / 16 - 1);
  const int b     = blockIdx.x >> 7;  // SEQ/16 == 128
  const int kbase = ktile * 16;

  const float* qb   = qb_all + (size_t)b * SEQ * D_HEAD;
  const float* krow = kb_all + (size_t)b * SEQ * D_HEAD +
                      (size_t)(kbase + col) * D_HEAD;

  float m = -INFINITY, l = 0.f;
  for (int qt = 0; qt < 16; ++qt) {
    const int q0 = w * 256 + qt * 16;
    const float* qrow = qb + (size_t)(q0 + col) * D_HEAD;
    v8f c = {};
#pragma unroll
    for (int kk = 0; kk < D_HEAD; kk += 4) {
      v2f a, bb;
      a.x  = qrow[kk + 2 * half];
      a.y  = qrow[kk + 2 * half + 1];
      bb.x = krow[kk + 2 * half];
      bb.y = krow[kk + 2 * half + 1];
      c = wmma_k4(a, bb, c);
    }
    float tm = c[0];
#pragma unroll
    for (int vv = 1; vv < 8; ++vv) tm = fmaxf(tm, c[vv]);
    float mn = fmaxf(m, tm);
    float s = 0.f;
#pragma unroll
    for (int vv = 0; vv < 8; ++vv) s += __expf(c[vv] - mn);
    l = l * __expf(m - mn) + s;
    m = mn;
  }
  // merge lane L with L+16 (same column, other 8 rows of each tile)
  float m2 = __shfl_xor(m, 16, 32);
  float l2 = __shfl_xor(l, 16, 32);
  float mn = fmaxf(m, m2);
  l = l * __expf(m - mn) + l2 * __expf(m2 - mn);
  m = mn;

  if (lane < 16) { sm[w][col] = m; sl[w][col] = l; }
  __syncthreads();
  if (threadIdx.x < 16) {
    float mm = sm[0][threadIdx.x], ll = sl[0][threadIdx.x];
#pragma unroll
    for (int ww = 1; ww < 8; ++ww) {
      float mo  = sm[ww][threadIdx.x];
      float mn2 = fmaxf(mm, mo);
      ll = ll * __expf(mm - mn2) + sl[ww][threadIdx.x] * __expf(mo - mn2);
      mm = mn2;
    }
    mOut[b * SEQ + kbase + threadIdx.x] = mm;
    lOut[b * SEQ + kbase + threadIdx.x] = ll;
  }
}

// ---------------------------------------------------------------------------
// Pass 2: ctx[q, :] = sum_k exp(s(q,k)-m[k])/l[k] * v[k, :].
// Block owns one 16-query tile; 8 waves split the 2048 keys; score tile is
// re-derived with WMMA, normalized, transposed through padded LDS into
// A-layout, then fed to 4 more WMMAs per K-step for the 16x64 ctx tile.
// Cross-wave partials reduced deterministically in LDS.
// ---------------------------------------------------------------------------
__global__ void k_ctx(const float* __restrict__ qb_all,
                      const float* __restrict__ kb_all,
                      const float* __restrict__ vb_all,
                      const float* __restrict__ mIn, const float* __restrict__ lIn,
                      float* __restrict__ ctx) {
  __shared__ float lds_p[8][16 * 17];   // stride 17: avoid 2-way bank conflict
  __shared__ float lds_c[8][16 * 64];
  const int w    = threadIdx.x >> 5;
  const int lane = threadIdx.x & 31;
  const int col  = lane & 15;
  const int half = lane >> 4;
  const int qbase = blockIdx.x * 16;        // global row
  const int b     = qbase / SEQ;
  const int qloc  = qbase - b * SEQ;

  const float* qb = qb_all + (size_t)b * SEQ * D_HEAD;
  const float* kb = kb_all + (size_t)b * SEQ * D_HEAD;
  const float* vb = vb_all + (size_t)b * SEQ * D_HEAD;
  const float* qrow = qb + (size_t)(qloc + col) * D_HEAD;

  v8f acc[4] = {{}, {}, {}, {}};
  for (int kt = 0; kt < 16; ++kt) {
    const int k0 = w * 256 + kt * 16;  // key row within batch
    const float* krow = kb + (size_t)(k0 + col) * D_HEAD;
    v8f c = {};
#pragma unroll
    for (int kk = 0; kk < D_HEAD; kk += 4) {
      v2f a, bb;
      a.x  = qrow[kk + 2 * half];
      a.y  = qrow[kk + 2 * half + 1];
      bb.x = krow[kk + 2 * half];
      bb.y = krow[kk + 2 * half + 1];
      c = wmma_k4(a, bb, c);
    }
    const float mc = mIn[b * SEQ + k0 + col];
    const float rl = 1.f / lIn[b * SEQ + k0 + col];
#pragma unroll
    for (int vv = 0; vv < 8; ++vv) {
      // element (q = vv+8*half, k = col) of the probability tile
      lds_p[w][(vv + 8 * half) * 17 + col] = __expf(c[vv] - mc) * rl;
    }
    // ctx += P(16x16) @ V(16x64), K in steps of 4 (wave-internal LDS, no barrier)
#pragma unroll
    for (int kk = 0; kk < 16; kk += 4) {
      v2f a;
      a.x = lds_p[w][col * 17 + kk + 2 * half];
      a.y = lds_p[w][col * 17 + kk + 2 * half + 1];
      const float* v0 = vb + (size_t)(k0 + kk + 2 * half) * D_HEAD + col;
      const float* v1 = v0 + D_HEAD;
#pragma unroll
      for (int t = 0; t < 4; ++t) {
        v2f bb;
        bb.x = v0[t * 16];
        bb.y = v1[t * 16];
        acc[t] = wmma_k4(a, bb, acc[t]);
      }
    }
  }
#pragma unroll
  for (int t = 0; t < 4; ++t)
#pragma unroll
    for (int vv = 0; vv < 8; ++vv)
      lds_c[w][(vv + 8 * half) * 64 + t * 16 + col] = acc[t][vv];
  __syncthreads();
#pragma unroll
  for (int r = 0; r < 4; ++r) {
    const int e = threadIdx.x + r * 256;  // qlocal*64 + d
    float s = 0.f;
#pragma unroll
    for (int ww = 0; ww < 8; ++ww) s += lds_c[ww][e];
    ctx[(size_t)(qbase + (e >> 6)) * D_HEAD + (e & 63)] = s;
  }
}

// ---------------------------------------------------------------------------
// out[n, j] = ctx[n, 0:64] @ Wsum[:, j] + bo[j]   (M=8192, N=1024, K=64)
// ---------------------------------------------------------------------------
__global__ void k_out(const float* __restrict__ ctx, const float* __restrict__ Wsum,
                      const float* __restrict__ bo, float* __restrict__ out) {
  const int w    = threadIdx.x >> 5;
  const int lane = threadIdx.x & 31;
  const int col  = lane & 15;
  const int half = lane >> 4;
  const int tile  = blockIdx.x * 8 + w;  // 0 .. 512*64-1
  const int mbase = (tile >> 6) * 16;
  const int nbase = (tile & 63) * 16;

  const float* crow = ctx + (size_t)(mbase + col) * D_HEAD;
  v8f acc = {};
#pragma unroll
  for (int d = 0; d < D_HEAD; d += 4) {
    v2f a, bb;
    a.x = crow[d + 2 * half];
    a.y = crow[d + 2 * half + 1];
    const float* w0 = Wsum + (size_t)(d + 2 * half) * D_MODEL + nbase + col;
    bb.x = w0[0];
    bb.y = w0[D_MODEL];
    acc = wmma_k4(a, bb, acc);
  }
  const float bbias = bo[nbase + col];
#pragma unroll
  for (int vv = 0; vv < 8; ++vv)
    out[(size_t)(mbase + vv + 8 * half) * D_MODEL + nbase + col] = acc[vv] + bbias;
}

extern "C" void kernel_launch(void* const* d_in, const int* in_sizes, int n_in,
                              void* d_out, int out_size, void* d_ws, size_t ws_size,
                              hipStream_t stream) {
  (void)in_sizes; (void)n_in; (void)out_size; (void)ws_size;
  const float* Q  = (const float*)d_in[0];
  const float* K  = (const float*)d_in[1];
  const float* V  = (const float*)d_in[2];
  const float* Wq = (const float*)d_in[3];
  const float* bq = (const float*)d_in[4];
  const float* Wk = (const float*)d_in[5];
  const float* bk = (const float*)d_in[6];
  const float* Wv = (const float*)d_in[7];
  const float* bv = (const float*)d_in[8];
  const float* Wo = (const float*)d_in[9];
  const float* bo = (const float*)d_in[10];
  float* out = (float*)d_out;

  float* ws   = (float*)d_ws;
  float* qbuf = ws;                       // 8192*64
  float* kbuf = qbuf + NROWS * D_HEAD;    // 8192*64
  float* vbuf = kbuf + NROWS * D_HEAD;    // 8192*64
  float* mbuf = vbuf + NROWS * D_HEAD;    // 8192
  float* lbuf = mbuf + NROWS;             // 8192
  float* cbuf = lbuf + NROWS;             // 8192*64
  float* wsum = cbuf + NROWS * D_HEAD;    // 64*1024   (~8.7 MB total)

  k_wsum<<<(D_HEAD * D_MODEL) / 256, 256, 0, stream>>>(Wo, wsum);
  k_proj<<<NROWS / 16 / 8, 256, 0, stream>>>(Q, Wq, bq, qbuf, 0.125f);
  k_proj<<<NROWS / 16 / 8, 256, 0, stream>>>(K, Wk, bk, kbuf, 1.0f);
  k_proj<<<NROWS / 16 / 8, 256, 0, stream>>>(V, Wv, bv, vbuf, 1.0f);
  k_stats<<<BATCH * (SEQ / 16), 256, 0, stream>>>(qbuf, kbuf, mbuf, lbuf);
  k_ctx<<<NROWS / 16, 256, 0, stream>>>(qbuf, kbuf, vbuf, mbuf, lbuf, cbuf);
  k_out<<<(NROWS / 16) * (D_MODEL / 16) / 8, 256, 0, stream>>>(cbuf, wsum, bo, out);
}